// SrgcnHead_25744033972467
// MI455X (gfx1250) — compile-verified
//
#include <hip/hip_runtime.h>
#include <hip/hip_bf16.h>

typedef __attribute__((ext_vector_type(2))) float v2f;
typedef __attribute__((ext_vector_type(8))) float v8f;

#define F_IN  128
#define F_OUT 64

// ---------------------------------------------------------------------------
// Zero-fill scratch (accumulator regions only)
// ---------------------------------------------------------------------------
__global__ void srgcn_zero(float* __restrict__ p, size_t n) {
    size_t i = (size_t)blockIdx.x * blockDim.x + threadIdx.x;
    size_t stride = (size_t)gridDim.x * blockDim.x;
    for (; i < n; i += stride) p[i] = 0.0f;
}

// ---------------------------------------------------------------------------
// GEMM: h0 = x @ W0, h1 = x @ W1  via V_WMMA_F32_16X16X4_F32
// block = 256 threads = 8 waves; wave w: hop = w&1, rowSub = w>>1
// Each wave computes the full 16x64 strip for one hop: 4 WMMA accumulators
// share a single A-tile load per K-step (4x arithmetic intensity on x).
// ---------------------------------------------------------------------------
__global__ void srgcn_gemm_wmma(const float* __restrict__ x,
                                const float* __restrict__ W0,
                                const float* __restrict__ W1,
                                float* __restrict__ h0,
                                float* __restrict__ h1,
                                int nRows) {
    const int wave = threadIdx.x >> 5;
    const int lane = threadIdx.x & 31;
    const int hop = wave & 1;
    const int rowSub = wave >> 1;
    const int rowBase = blockIdx.x * 64 + rowSub * 16;
    if (rowBase >= nRows) return;          // wave-uniform exit

    const float* __restrict__ W = hop ? W1 : W0;
    float* __restrict__ H = hop ? h1 : h0;

    const int halfsel = lane >> 4;         // 0: lanes 0-15, 1: lanes 16-31
    const int mLane = lane & 15;

    // Clamp row for loads so EXEC stays all-ones through the WMMA loop.
    int row = rowBase + mLane;
    int rowClamped = row < nRows ? row : (nRows - 1);
    const float* __restrict__ xrow = x + (size_t)rowClamped * F_IN;

    v8f acc0 = {}, acc1 = {}, acc2 = {}, acc3 = {};
    for (int k = 0; k < F_IN; k += 4) {
        // A (16x4 f32): VGPR0 = K=k+2*halfsel, VGPR1 = K=k+2*halfsel+1
        v2f a;
        a.x = xrow[k + 2 * halfsel];
        a.y = xrow[k + 2 * halfsel + 1];
        // B (4x16 f32): VGPR0 holds K=k (lanes0-15) / K=k+2 (lanes16-31),
        //               VGPR1 holds K=k+1 / K=k+3; 4 column tiles
        const float* __restrict__ wr0 = W + (size_t)(k + 2 * halfsel) * F_OUT + mLane;
        const float* __restrict__ wr1 = wr0 + F_OUT;
        v2f b0v = { wr0[0],  wr1[0]  };
        v2f b1v = { wr0[16], wr1[16] };
        v2f b2v = { wr0[32], wr1[32] };
        v2f b3v = { wr0[48], wr1[48] };
        acc0 = __builtin_amdgcn_wmma_f32_16x16x4_f32(false, a, false, b0v, (short)0, acc0, false, false);
        acc1 = __builtin_amdgcn_wmma_f32_16x16x4_f32(false, a, false, b1v, (short)0, acc1, false, false);
        acc2 = __builtin_amdgcn_wmma_f32_16x16x4_f32(false, a, false, b2v, (short)0, acc2, false, false);
        acc3 = __builtin_amdgcn_wmma_f32_16x16x4_f32(false, a, false, b3v, (short)0, acc3, false, false);
    }

    // C/D layout: VGPR r -> M = r + 8*halfsel, N = mLane + {0,16,32,48}
    const bool fullTile = (rowBase + 16) <= nRows;   // wave-uniform
    if (fullTile) {
        #pragma unroll
        for (int r = 0; r < 8; ++r) {
            int m = rowBase + r + 8 * halfsel;
            float* dst = H + (size_t)m * F_OUT + mLane;
            dst[0]  = acc0[r];
            dst[16] = acc1[r];
            dst[32] = acc2[r];
            dst[48] = acc3[r];
        }
    } else {
        #pragma unroll
        for (int r = 0; r < 8; ++r) {
            int m = rowBase + r + 8 * halfsel;
            if (m < nRows) {
                float* dst = H + (size_t)m * F_OUT + mLane;
                dst[0]  = acc0[r];
                dst[16] = acc1[r];
                dst[32] = acc2[r];
                dst[48] = acc3[r];
            }
        }
    }
}

// ---------------------------------------------------------------------------
// Degree passes: deg[r] += val ; deg2[r] += val * deg[c]
// ---------------------------------------------------------------------------
__global__ void srgcn_deg1(const int* __restrict__ row,
                           const float* __restrict__ val,
                           float* __restrict__ deg, int E) {
    int e = blockIdx.x * blockDim.x + threadIdx.x;
    if (e < E) unsafeAtomicAdd(&deg[row[e]], val[e]);
}

__global__ void srgcn_deg2(const int* __restrict__ row,
                           const int* __restrict__ col,
                           const float* __restrict__ val,
                           const float* __restrict__ deg,
                           float* __restrict__ deg2, int E) {
    int e = blockIdx.x * blockDim.x + threadIdx.x;
    if (e < E) unsafeAtomicAdd(&deg2[row[e]], val[e] * deg[col[e]]);
}

// ---------------------------------------------------------------------------
// Feature SpMM: out[row[e], :] += val[e] * v[col[e], :]
// 64 threads per edge, thread f handles feature f (fully coalesced 256B rows)
// ---------------------------------------------------------------------------
__global__ void srgcn_spmm64(const int* __restrict__ row,
                             const int* __restrict__ col,
                             const float* __restrict__ val,
                             const float* __restrict__ v,
                             float* __restrict__ out, int E) {
    long long t = (long long)blockIdx.x * blockDim.x + threadIdx.x;
    int e = (int)(t >> 6);
    int f = (int)(t & 63);
    if (e < E) {
        float contrib = val[e] * v[(size_t)col[e] * F_OUT + f];
        unsafeAtomicAdd(&out[(size_t)row[e] * F_OUT + f], contrib);
    }
}

// ---------------------------------------------------------------------------
// Finalize: y = mask(agg/deg) + b ; out = relu(y) + sigmoid(y.fc+bf)*min(y,0)
// one wave32 per (node, hop); each lane handles features lane and lane+32
// ---------------------------------------------------------------------------
__global__ void srgcn_finalize(const float* __restrict__ agg0,
                               const float* __restrict__ deg,
                               const float* __restrict__ agg1,
                               const float* __restrict__ deg2,
                               const float* __restrict__ b0,
                               const float* __restrict__ b1,
                               const float* __restrict__ fc0,
                               const float* __restrict__ bf0,
                               const float* __restrict__ fc1,
                               const float* __restrict__ bf1,
                               float* __restrict__ out, int N) {
    const int wave = threadIdx.x >> 5;
    const int lane = threadIdx.x & 31;
    const int task = blockIdx.x * 8 + wave;   // 8 waves per 256-thread block
    const int node = task >> 1;
    const int hop = task & 1;
    if (node >= N) return;

    const float* __restrict__ agg = hop ? agg1 : agg0;
    const float* __restrict__ dg  = hop ? deg2 : deg;
    const float* __restrict__ b   = hop ? b1 : b0;
    const float* __restrict__ fc  = hop ? fc1 : fc0;
    const float bf = hop ? bf1[0] : bf0[0];

    const float d = dg[node];
    const float inv = d > 0.0f ? 1.0f / d : 0.0f;
    const size_t base = (size_t)node * F_OUT;

    float y0 = (d > 0.0f ? agg[base + lane] * inv : 0.0f) + b[lane];
    float y1 = (d > 0.0f ? agg[base + lane + 32] * inv : 0.0f) + b[lane + 32];

    float part = y0 * fc[lane] + y1 * fc[lane + 32];
    #pragma unroll
    for (int off = 16; off >= 1; off >>= 1)
        part += __shfl_xor(part, off, 32);

    float g = 1.0f / (1.0f + __expf(-(part + bf)));
    float e0 = fmaxf(y0, 0.0f) + g * fminf(y0, 0.0f);
    float e1 = fmaxf(y1, 0.0f) + g * fminf(y1, 0.0f);

    out[(size_t)node * (2 * F_OUT) + hop * F_OUT + lane]      = e0;
    out[(size_t)node * (2 * F_OUT) + hop * F_OUT + lane + 32] = e1;
}

// ---------------------------------------------------------------------------
// Launch
// ---------------------------------------------------------------------------
extern "C" void kernel_launch(void* const* d_in, const int* in_sizes, int n_in,
                              void* d_out, int out_size, void* d_ws, size_t ws_size,
                              hipStream_t stream) {
    const float* x         = (const float*)d_in[0];
    const int*   eidx      = (const int*)d_in[1];
    const float* edge_attr = (const float*)d_in[2];
    const float* W0        = (const float*)d_in[3];
    const float* b0        = (const float*)d_in[4];
    const float* W1        = (const float*)d_in[5];
    const float* b1        = (const float*)d_in[6];
    const float* fc0       = (const float*)d_in[7];
    const float* bf0       = (const float*)d_in[8];
    const float* fc1       = (const float*)d_in[9];
    const float* bf1       = (const float*)d_in[10];
    float* out = (float*)d_out;

    const int N = in_sizes[0] / F_IN;
    const int E = in_sizes[2];
    const int* row = eidx;
    const int* col = eidx + E;

    // Workspace layout (floats)
    float* ws    = (float*)d_ws;
    size_t nh    = (size_t)N * F_OUT;
    float* h0    = ws;
    float* h1    = ws + nh;
    float* agg0  = ws + 2 * nh;
    float* agg1a = ws + 3 * nh;
    float* agg1b = ws + 4 * nh;
    float* deg   = ws + 5 * nh;
    float* deg2  = ws + 5 * nh + N;

    // 1) zero the accumulated regions (agg0, agg1a, agg1b, deg, deg2)
    size_t nzero = 3 * nh + 2 * (size_t)N;
    srgcn_zero<<<2048, 256, 0, stream>>>(agg0, nzero);

    // 2) GEMMs via f32 WMMA (64 rows per block: 4 row sub-tiles x 2 hops)
    int rowBlocks = (N + 63) / 64;
    srgcn_gemm_wmma<<<rowBlocks, 256, 0, stream>>>(x, W0, W1, h0, h1, N);

    // 3) degree passes
    int eb = (E + 255) / 256;
    srgcn_deg1<<<eb, 256, 0, stream>>>(row, edge_attr, deg, E);
    srgcn_deg2<<<eb, 256, 0, stream>>>(row, col, edge_attr, deg, deg2, E);

    // 4) feature SpMMs
    long long spmmThreads = (long long)E * F_OUT;
    int sb = (int)((spmmThreads + 255) / 256);
    srgcn_spmm64<<<sb, 256, 0, stream>>>(row, col, edge_attr, h0, agg0, E);
    srgcn_spmm64<<<sb, 256, 0, stream>>>(row, col, edge_attr, h1, agg1a, E);
    srgcn_spmm64<<<sb, 256, 0, stream>>>(row, col, edge_attr, agg1a, agg1b, E);

    // 5) epilogue
    int tasks = 2 * N;
    int fb = (tasks + 7) / 8;
    srgcn_finalize<<<fb, 256, 0, stream>>>(agg0, deg, agg1b, deg2,
                                           b0, b1, fc0, bf0, fc1, bf1, out, N);
}